// QFeatureMapOneHot_1176821039162
// MI455X (gfx1250) — compile-verified
//
#include <hip/hip_runtime.h>
#include <hip/hip_bf16.h>

typedef __attribute__((ext_vector_type(2))) float v2f;
typedef __attribute__((ext_vector_type(4))) float v4f;
typedef __attribute__((ext_vector_type(8))) float v8f;

#define BATCH 256
#define DIM   20
#define ROWLEN (1 << DIM)   // 1048576 floats per row

// ---------------------------------------------------------------------------
// Kernel 1: idx[r] = sum_j in[r*20+j] << (19-j), computed as a 256x20 mat-vec
// on the WMMA pipe: D = A x B, A[m][k] = bit * 2^(19-k) (radix folded into A,
// exact in fp32 since idx < 2^20 << 2^24), B = all-ones (lane-layout proof).
// One wave (32 threads), 16 M-tiles x 5 K-steps = 80 v_wmma_f32_16x16x4_f32.
// A layout (ISA 7.12.2, 32-bit A 16x4): lane 0-15 -> K=0(v0),1(v1);
// lane 16-31 -> K=2(v0),3(v1); M = lane&15.
// D layout (32-bit C/D 16x16): lanes 0-15 VGPR j -> M=j; lanes 16-31 -> M=8+j.
// ---------------------------------------------------------------------------
__global__ __launch_bounds__(32) void onehot_idx_wmma(const int* __restrict__ in,
                                                      int* __restrict__ idx) {
    const int lane = threadIdx.x;        // 0..31, wave32
    const int half = lane >> 4;          // 0: K pair {0,1}; 1: K pair {2,3}
    const int m    = lane & 15;

    v2f b;                               // all-ones B: layout-independent
    b.x = 1.0f; b.y = 1.0f;

    for (int mt = 0; mt < 16; ++mt) {
        const int row = mt * 16 + m;
        v8f c = {};
        #pragma unroll
        for (int kk = 0; kk < 5; ++kk) {
            const int k0 = kk * 4 + half * 2;       // 0..18, k0+1 <= 19
            const int bit0 = in[row * DIM + k0];
            const int bit1 = in[row * DIM + k0 + 1];
            v2f a;
            a.x = (float)(bit0 << (DIM - 1 - k0));
            a.y = (float)(bit1 << (DIM - 1 - (k0 + 1)));
            c = __builtin_amdgcn_wmma_f32_16x16x4_f32(
                    /*neg_a=*/false, a, /*neg_b=*/false, b,
                    /*c_mod=*/(short)0, c, /*reuse_a=*/false, /*reuse_b=*/false);
        }
        // All N columns of D are equal. Lane 0 holds M=0..7, lane 16 holds M=8..15.
        if ((lane & 15) == 0) {
            #pragma unroll
            for (int j = 0; j < 8; ++j)
                idx[mt * 16 + half * 8 + j] = (int)c[j];
        }
    }
}

// ---------------------------------------------------------------------------
// Kernel 2: the bandwidth kernel. Writes all 2^20 floats of row blockIdx.y,
// 1.0f at column idx[row], 0.0f elsewhere. 16B non-temporal stores (the 1 GB
// stream would otherwise thrash the 192 MB L2). 64 B per thread.
// grid = (256, 256), block = 256 (8 wave32s).
// ---------------------------------------------------------------------------
__global__ __launch_bounds__(256) void onehot_fill(const int* __restrict__ idx,
                                                   float* __restrict__ out) {
    const int row = blockIdx.y;
    const int hot = idx[row];                        // uniform -> s_load
    float* __restrict__ rowp = out + ((size_t)row << DIM);

    const int chunk0 = blockIdx.x * (256 * 4) + threadIdx.x;  // float4 chunks
    #pragma unroll
    for (int i = 0; i < 4; ++i) {
        const int chunk = chunk0 + i * 256;
        const int col = chunk << 2;
        v4f v;
        v.x = (col + 0 == hot) ? 1.0f : 0.0f;
        v.y = (col + 1 == hot) ? 1.0f : 0.0f;
        v.z = (col + 2 == hot) ? 1.0f : 0.0f;
        v.w = (col + 3 == hot) ? 1.0f : 0.0f;
        __builtin_nontemporal_store(v, (v4f*)(rowp + ((size_t)chunk << 2)));
    }
}

extern "C" void kernel_launch(void* const* d_in, const int* in_sizes, int n_in,
                              void* d_out, int out_size, void* d_ws, size_t ws_size,
                              hipStream_t stream) {
    const int* in  = (const int*)d_in[0];   // (256, 20) int32, values in {0,1}
    int*       idx = (int*)d_ws;            // 256 ints of scratch
    float*     out = (float*)d_out;         // (256, 2^20) fp32

    hipLaunchKernelGGL(onehot_idx_wmma, dim3(1), dim3(32), 0, stream, in, idx);

    // 2^20 floats / row = 262144 float4 chunks; 1024 chunks per block.
    hipLaunchKernelGGL(onehot_fill, dim3(256, BATCH), dim3(256), 0, stream, idx, out);
}